// ANN_Net_146028888292
// MI455X (gfx1250) — compile-verified
//
#include <hip/hip_runtime.h>
#include <hip/hip_bf16.h>

// MI455X (gfx1250): wave32, WMMA. This workload is HBM-bound (~6 FLOP/byte),
// so we stream x_in once and run the 13 tiny FP32 GEMVs on the matrix pipe
// with V_WMMA_F32_16X16X4_F32 in a transposed formulation (D = W^T * X^T + b),
// keeping all activations in registers. Layer-to-layer layout fixup is a
// single half-wave swap via ds_swizzle SWAPX16 (no LDS memory traffic).

typedef __attribute__((ext_vector_type(2))) float v2f;
typedef __attribute__((ext_vector_type(8))) float v8f;

#define RRELU_SLOPE 0.0025f

__device__ __forceinline__ v8f wmma4(v2f a, v2f b, v8f c) {
  // D(16x16,f32) = A(16x4,f32) * B(4x16,f32) + C
  return __builtin_amdgcn_wmma_f32_16x16x4_f32(
      /*neg_a=*/false, a, /*neg_b=*/false, b,
      /*c_mod=*/(short)0, c, /*reuse_a=*/false, /*reuse_b=*/false);
}

// Bring lane (l^16)'s value to lane l (half-wave swap). Group-of-32 mode,
// xor_mask=0x10, and_mask=0x1f -> offset 0x401F.
__device__ __forceinline__ float swapx16(float v) {
  int i = __builtin_amdgcn_ds_swizzle(__builtin_bit_cast(int, v), 0x401f);
  return __builtin_bit_cast(float, i);
}

__device__ __forceinline__ v8f rrelu8(v8f x) {
  v8f r;
#pragma unroll
  for (int i = 0; i < 8; ++i) r[i] = (x[i] >= 0.0f) ? x[i] : RRELU_SLOPE * x[i];
  return r;
}

// Activation D-layout: component i of lane l = feature (i + (l>>4)*8), row (l&15).
// B-operand layout (4x16 f32): component v of lane l = K (v + (l>>4)*2), N (l&15).
// Lower half-wave reads features KO..KO+1 directly; upper half-wave needs
// features KO+2..KO+3 which live in the lower half-wave -> ds_swizzle swap.
template <int KO>
__device__ __forceinline__ v2f make_b(const v8f& d, int half) {
  float s0 = swapx16(d[KO + 2]);
  float s1 = swapx16(d[KO + 3]);
  v2f b;
  b.x = half ? s0 : d[KO + 0];
  b.y = half ? s1 : d[KO + 1];
  return b;
}

// A-operand (16x4 f32): component v of lane l = W^T[M = l&15][K = KO + (l>>4)*2 + v],
// zero-padded outside (FI x FO). W is row-major (FI rows, FO cols).
template <int FI, int FO, int KO>
__device__ __forceinline__ v2f make_a(const float* W, int m, int half) {
  int k0 = KO + half * 2;
  int k1 = k0 + 1;
  int mc  = (m  < FO) ? m  : (FO - 1);
  int k0c = (k0 < FI) ? k0 : (FI - 1);
  int k1c = (k1 < FI) ? k1 : (FI - 1);
  float w0 = W[k0c * FO + mc];
  float w1 = W[k1c * FO + mc];
  v2f a;
  a.x = (m < FO && k0 < FI) ? w0 : 0.0f;
  a.y = (m < FO && k1 < FI) ? w1 : 0.0f;
  return a;
}

// Bias as C accumulator: component i of lane l = bias[i + (l>>4)*8], padded with 0.
template <int FO>
__device__ __forceinline__ v8f make_c(const float* b, int half) {
  v8f c;
#pragma unroll
  for (int i = 0; i < 8; ++i) {
    float bv = (i < FO) ? b[i] : 0.0f;  // i<FO is compile-time; FO<=8
    c[i] = half ? 0.0f : bv;
  }
  return c;
}

__device__ __forceinline__ v8f lin4(const v8f& x, v2f a, const v8f& c, int half) {
  return wmma4(a, make_b<0>(x, half), c);
}
__device__ __forceinline__ v8f lin8(const v8f& x, v2f a0, v2f a1, const v8f& c, int half) {
  v8f d = wmma4(a0, make_b<0>(x, half), c);
  return wmma4(a1, make_b<4>(x, half), d);
}

struct Params {
  const float* x;
  const float *W_in, *b_in, *W_h1, *b_h1, *W_h2, *b_h2, *W_h3, *b_h3;
  const float *W_h4, *b_h4, *W_h5, *b_h5, *W_en, *b_en, *W_h6, *b_h6;
  const float *W_h7, *b_h7, *W_h8, *b_h8, *W_h9, *b_h9, *W_h10, *b_h10;
  const float *W_de, *b_de;
  float* enc;
  float* dec;
  int ntiles;
};

__global__ __launch_bounds__(256) void ann_autoenc_wmma(Params p) {
  const int lane = threadIdx.x & 31;
  const int half = lane >> 4;
  const int m    = lane & 15;
  const int kb   = half * 2;

  // Hoist all weight fragments (tiny, cache-resident; loop-invariant).
  const v2f A_in0 = make_a<8, 8, 0>(p.W_in, m, half);
  const v2f A_in1 = make_a<8, 8, 4>(p.W_in, m, half);
  const v8f C_in  = make_c<8>(p.b_in, half);
  const v2f A_h1a = make_a<8, 4, 0>(p.W_h1, m, half);
  const v2f A_h1b = make_a<8, 4, 4>(p.W_h1, m, half);
  const v8f C_h1  = make_c<4>(p.b_h1, half);
  const v2f A_h2 = make_a<4, 4, 0>(p.W_h2, m, half);  const v8f C_h2 = make_c<4>(p.b_h2, half);
  const v2f A_h3 = make_a<4, 4, 0>(p.W_h3, m, half);  const v8f C_h3 = make_c<4>(p.b_h3, half);
  const v2f A_h4 = make_a<4, 4, 0>(p.W_h4, m, half);  const v8f C_h4 = make_c<4>(p.b_h4, half);
  const v2f A_h5 = make_a<4, 4, 0>(p.W_h5, m, half);  const v8f C_h5 = make_c<4>(p.b_h5, half);
  const v2f A_en = make_a<4, 1, 0>(p.W_en, m, half);  const v8f C_en = make_c<1>(p.b_en, half);
  const v2f A_h6 = make_a<1, 4, 0>(p.W_h6, m, half);  const v8f C_h6 = make_c<4>(p.b_h6, half);
  const v2f A_h7 = make_a<4, 4, 0>(p.W_h7, m, half);  const v8f C_h7 = make_c<4>(p.b_h7, half);
  const v2f A_h8 = make_a<4, 4, 0>(p.W_h8, m, half);  const v8f C_h8 = make_c<4>(p.b_h8, half);
  const v2f A_h9 = make_a<4, 4, 0>(p.W_h9, m, half);  const v8f C_h9 = make_c<4>(p.b_h9, half);
  const v2f A_hA = make_a<4, 4, 0>(p.W_h10, m, half); const v8f C_hA = make_c<4>(p.b_h10, half);
  const v2f A_de = make_a<4, 8, 0>(p.W_de, m, half);  const v8f C_de = make_c<8>(p.b_de, half);

  const int wavesPerBlock = blockDim.x >> 5;
  const int wave   = blockIdx.x * wavesPerBlock + (threadIdx.x >> 5);
  const int nwaves = gridDim.x * wavesPerBlock;

  for (int tile = wave; tile < p.ntiles; tile += nwaves) {
    const size_t row = (size_t)tile * 16 + m;   // N-index of B = row (l&15)
    const float* xr = p.x + row * 8 + kb;       // features kb..kb+1 / kb+4..kb+5

    __builtin_prefetch(xr + (size_t)nwaves * 128, 0, 0);  // next grid-stride tile

    v2f bx0 = *(const v2f*)(xr);      // X^T B-fragment, K = 0..3
    v2f bx1 = *(const v2f*)(xr + 4);  // X^T B-fragment, K = 4..7

    // in: 8 -> 8 (two K=4 WMMAs, accumulate chain)
    v8f t = wmma4(A_in0, bx0, C_in);
    t = wmma4(A_in1, bx1, t);
    t = rrelu8(t);
    // h1: 8 -> 4
    v8f x4 = lin8(t, A_h1a, A_h1b, C_h1, half);

    // residual block 1: x += h3(rrelu(h2(x)))
    v8f u = rrelu8(lin4(x4, A_h2, C_h2, half));
    x4 = x4 + lin4(u, A_h3, C_h3, half);
    // residual block 2
    u = rrelu8(lin4(x4, A_h4, C_h4, half));
    x4 = x4 + lin4(u, A_h5, C_h5, half);

    // encode: 4 -> 1
    v8f xe = rrelu8(lin4(x4, A_en, C_en, half));

    // h6: 1 -> 4 (padded slots of xe are exactly 0, generic path is correct)
    v8f y = rrelu8(lin4(xe, A_h6, C_h6, half));
    // residual block 3
    u = rrelu8(lin4(y, A_h7, C_h7, half));
    y = y + lin4(u, A_h8, C_h8, half);
    // residual block 4
    u = rrelu8(lin4(y, A_h9, C_h9, half));
    y = y + lin4(u, A_hA, C_hA, half);

    // decode: 4 -> 8
    v8f xd = rrelu8(lin4(y, A_de, C_de, half));

    // Lower half-wave holds all outputs: feature i of row (l&15) in component i.
    if (!half) {
      p.enc[row] = xe[0];
      *(v8f*)(p.dec + row * 8) = xd;  // 32B contiguous per row -> 2x b128
    }
  }
}

extern "C" void kernel_launch(void* const* d_in, const int* in_sizes, int n_in,
                              void* d_out, int out_size, void* d_ws, size_t ws_size,
                              hipStream_t stream) {
  (void)n_in; (void)d_ws; (void)ws_size; (void)out_size;
  Params p;
  p.x    = (const float*)d_in[0];
  p.W_in = (const float*)d_in[1];  p.b_in  = (const float*)d_in[2];
  p.W_h1 = (const float*)d_in[3];  p.b_h1  = (const float*)d_in[4];
  p.W_h2 = (const float*)d_in[5];  p.b_h2  = (const float*)d_in[6];
  p.W_h3 = (const float*)d_in[7];  p.b_h3  = (const float*)d_in[8];
  p.W_h4 = (const float*)d_in[9];  p.b_h4  = (const float*)d_in[10];
  p.W_h5 = (const float*)d_in[11]; p.b_h5  = (const float*)d_in[12];
  p.W_en = (const float*)d_in[13]; p.b_en  = (const float*)d_in[14];
  p.W_h6 = (const float*)d_in[15]; p.b_h6  = (const float*)d_in[16];
  p.W_h7 = (const float*)d_in[17]; p.b_h7  = (const float*)d_in[18];
  p.W_h8 = (const float*)d_in[19]; p.b_h8  = (const float*)d_in[20];
  p.W_h9 = (const float*)d_in[21]; p.b_h9  = (const float*)d_in[22];
  p.W_h10= (const float*)d_in[23]; p.b_h10 = (const float*)d_in[24];
  p.W_de = (const float*)d_in[25]; p.b_de  = (const float*)d_in[26];

  const long rows = in_sizes[0] / 8;     // 4,194,304
  p.enc = (float*)d_out;                 // outputs concatenated: encode, then decode
  p.dec = (float*)d_out + rows;
  p.ntiles = (int)(rows / 16);           // rows % 16 == 0 for this workload

  const int threads = 256;               // 8 waves/block
  const int wavesPerBlock = threads / 32;
  int targetWaves = (p.ntiles + 3) / 4;  // ~4 tiles per wave (grid-stride)
  int blocks = (targetWaves + wavesPerBlock - 1) / wavesPerBlock;
  if (blocks < 1) blocks = 1;

  ann_autoenc_wmma<<<blocks, threads, 0, stream>>>(p);
}